// Glo_skipAttention_49520972923075
// MI455X (gfx1250) — compile-verified
//
#include <hip/hip_runtime.h>
#include <math.h>

// ---------------- problem constants ----------------
#define BB 4
#define CC 256
#define NHH 8
#define DHH 32
#define HWW 1024
#define NKV 1426
#define NKV_PAD 1440
#define SROW 1444          // LDS row stride (mod 64 == 36 -> conflict-free)
#define JTILES 90          // 1440/16
#define QTILES 65          // ceil(1025/16)
#define SCALE_F 0.17677669529663689f   // 32^-0.5
#define GLOSCALE_F 0.0625f             // 256^-0.5

typedef float v2f __attribute__((ext_vector_type(2)));
typedef float v4f __attribute__((ext_vector_type(4)));
typedef float v8f __attribute__((ext_vector_type(8)));

__device__ __forceinline__ v8f wmma4(v2f a, v2f b, v8f c) {
    // D(16x16,f32) = A(16x4,f32) x B(4x16,f32) + C
    return __builtin_amdgcn_wmma_f32_16x16x4_f32(false, a, false, b, (short)0, c,
                                                 false, false);
}

// ---------------- workspace layout (floats) ----------------
#define OFF_QVH   ((size_t)0)          // 4*512*1024           = 2,097,152
#define OFF_QL    ((size_t)2097152)    // 4*256*1024           = 1,048,576
#define OFF_PQ    ((size_t)3145728)    // 4*256*1426           = 1,460,224
#define OFF_PV    ((size_t)4605952)    //                        1,460,224
#define OFF_MSQ   ((size_t)6066176)    // 4*1440*256 (padded)  = 1,474,560
#define OFF_MSV   ((size_t)7540736)    // 4*1426*256 (pre-LN)  = 1,460,224
#define OFF_MSVT  ((size_t)9000960)    // 4*256*1440 (V^T pad) = 1,474,560
#define OFF_WATTN ((size_t)10475520)   // 8*32*32              = 8,192
// att_fea reuses OFF_PQ, out_img reuses OFF_PV, dwconv tmp reuses OFF_QL

// ---------------- generic NT-GEMM: out[b,o,p] = sum_k w[o,k]*x[b,k,p] ----------------
__global__ __launch_bounds__(256) void proj_kernel(const float* __restrict__ w,
                                                   const float* __restrict__ x,
                                                   float* __restrict__ out, int rows) {
    int idx = blockIdx.x * 256 + threadIdx.x;
    int p = idx & 1023;
    int rest = idx >> 10;
    int o = rest % rows;
    int b = rest / rows;
    const float* wr = w + (size_t)o * CC;
    const float* xp = x + ((size_t)b * CC) * HWW + p;
    float s = 0.f;
#pragma unroll 4
    for (int k = 0; k < CC; ++k) s += wr[k] * xp[(size_t)k * HWW];
    out[idx] = s;
}

// ---------------- multiscale average pooling ----------------
__device__ __forceinline__ void decode_tok(int t, int& s, int& off, int& nout) {
    if (t < 1024)      { s = 0; off = 0;    nout = 32; }
    else if (t < 1280) { s = 1; off = 1024; nout = 16; }
    else if (t < 1401) { s = 2; off = 1280; nout = 11; }
    else               { s = 3; off = 1401; nout = 5;  }
}

__global__ __launch_bounds__(256) void pool_kernel(const float* __restrict__ qvh,
                                                   float* __restrict__ pq,
                                                   float* __restrict__ pv) {
    int idx = blockIdx.x * 256 + threadIdx.x;
    if (idx >= 2 * BB * CC * NKV) return;
    int t = idx % NKV; int rest = idx / NKV;
    int c = rest % CC; rest /= CC;
    int b = rest % BB; int src = rest / BB;
    const float* x = qvh + (((size_t)b * 512) + src * 256 + c) * HWW;
    int s, off, nout; decode_tok(t, s, off, nout);
    int rel = t - off;
    int o = rel / nout, pp = rel % nout;
    int sh = (o * 32) / nout,      eh = ((o + 1) * 32 + nout - 1) / nout;
    int sw = (pp * 32) / nout,     ew = ((pp + 1) * 32 + nout - 1) / nout;
    float acc = 0.f;
    for (int hh = sh; hh < eh; ++hh)
        for (int ww = sw; ww < ew; ++ww) acc += x[hh * 32 + ww];
    acc /= (float)((eh - sh) * (ew - sw));
    float* dst = src ? pv : pq;
    dst[((size_t)b * CC + c) * NKV + t] = acc;
}

// residual depthwise 3x3 + bias; K side -> (b, 1440-padded t, c); V side -> (b, t, c)
__global__ __launch_bounds__(256) void convms_kernel(const float* __restrict__ pq,
                                                     const float* __restrict__ pv,
                                                     const float* __restrict__ dw,
                                                     const float* __restrict__ db,
                                                     float* __restrict__ msq,
                                                     float* __restrict__ msv) {
    int idx = blockIdx.x * 256 + threadIdx.x;
    if (idx >= 2 * BB * CC * NKV) return;
    int t = idx % NKV; int rest = idx / NKV;
    int c = rest % CC; rest /= CC;
    int b = rest % BB; int which = rest / BB;
    const float* src = which ? pv : pq;
    size_t base = ((size_t)b * CC + c) * NKV;
    int s, off, nout; decode_tok(t, s, off, nout);
    int rel = t - off;
    int o = rel / nout, pp = rel % nout;
    const float* wk = dw + ((size_t)s * CC + c) * 9;
    float acc = src[base + t] + db[s * CC + c];
    for (int dy = -1; dy <= 1; ++dy)
        for (int dx = -1; dx <= 1; ++dx) {
            int oo = o + dy, qq = pp + dx;
            if (oo >= 0 && oo < nout && qq >= 0 && qq < nout)
                acc += wk[(dy + 1) * 3 + (dx + 1)] * src[base + off + oo * nout + qq];
        }
    if (which) msv[((size_t)b * NKV + t) * CC + c] = acc;
    else       msq[((size_t)b * NKV_PAD + t) * CC + c] = acc;
}

// LayerNorm over C per token. K side in place; V side written transposed (d-major).
__global__ __launch_bounds__(256) void ln_kernel(float* __restrict__ msq,
                                                 const float* __restrict__ msv,
                                                 float* __restrict__ msvt,
                                                 const float* __restrict__ g,
                                                 const float* __restrict__ bta) {
    __shared__ float red[256];
    int blk = blockIdx.x;
    int which = blk & 1; int bt = blk >> 1;
    int b = bt / NKV, t = bt % NKV;
    int c = threadIdx.x;
    float x = which ? msv[((size_t)b * NKV + t) * CC + c]
                    : msq[((size_t)b * NKV_PAD + t) * CC + c];
    red[c] = x; __syncthreads();
    for (int s = 128; s > 0; s >>= 1) { if (c < s) red[c] += red[c + s]; __syncthreads(); }
    float mu = red[0] * (1.f / 256.f); __syncthreads();
    float d = x - mu;
    red[c] = d * d; __syncthreads();
    for (int s = 128; s > 0; s >>= 1) { if (c < s) red[c] += red[c + s]; __syncthreads(); }
    float var = red[0] * (1.f / 256.f);
    float y = d * rsqrtf(var + 1e-5f) * g[c] + bta[c];
    if (which) msvt[((size_t)b * CC + c) * NKV_PAD + t] = y;
    else       msq[((size_t)b * NKV_PAD + t) * CC + c] = y;
}

// zero the padded tokens (1426..1439) of K rows and V^T columns
__global__ __launch_bounds__(256) void pad_kernel(float* __restrict__ msq,
                                                  float* __restrict__ msvt) {
    int idx = blockIdx.x * 256 + threadIdx.x;
    if (idx >= BB * 14 * CC) return;
    int c = idx & 255; int r = idx >> 8;
    int t = NKV + (r % 14); int b = r / 14;
    msq[((size_t)b * NKV_PAD + t) * CC + c] = 0.f;
    msvt[((size_t)b * CC + c) * NKV_PAD + t] = 0.f;
}

// constant w_attn[h,i,j] from hl_ln_b / hl_w / hl_b (LN over size-1 axis collapses)
__global__ __launch_bounds__(512) void wattn_kernel(const float* __restrict__ hl_ln_b,
                                                    const float* __restrict__ hl_w,
                                                    const float* __restrict__ hl_b,
                                                    float* __restrict__ wattn) {
    __shared__ float val[512];
    int t = threadIdx.x;
    float xb = hl_ln_b[0];
    float g0 = 0.5f * xb * (1.0f + erff(xb * 0.7071067811865475f));  // exact gelu
    float s = 0.f;
    const float* row = hl_w + (size_t)t * CC;
    for (int c = 0; c < CC; ++c) s += row[c];
    val[t] = g0 * s + hl_b[t];
    __syncthreads();
    if (t < 256) {
        int h = t >> 5, i = t & 31;
        float wq = val[h * 64 + i] * GLOSCALE_F;
        float sc[32]; float mx = -3.0e38f;
        for (int j = 0; j < 32; ++j) { sc[j] = wq * val[h * 64 + 32 + j]; mx = fmaxf(mx, sc[j]); }
        float sum = 0.f;
        for (int j = 0; j < 32; ++j) { sc[j] = expf(sc[j] - mx); sum += sc[j]; }
        float inv = 1.f / sum;
        for (int j = 0; j < 32; ++j) wattn[(h * 32 + i) * 32 + j] = sc[j] * inv;
    }
}

// ---------------- fused attention: one wave per (b,h,16-query strip) ----------------
// k-slot permutation d = k*8 + ks makes every lane's fragment data contiguous (b128).
__global__ __launch_bounds__(32) void attn_kernel(const float* __restrict__ ql,
                                                  const float* __restrict__ glob_l,
                                                  const float* __restrict__ msq,
                                                  const float* __restrict__ msvt,
                                                  float* __restrict__ att_fea) {
    extern __shared__ float smem[];
    float* sS = smem;               // 16 * SROW
    float* sDen = smem + 16 * SROW; // 16
    int blk = blockIdx.x;
    int qt = blk % QTILES; int bh = blk / QTILES;
    int h = bh % NHH; int b = bh / NHH;
    int lane = threadIdx.x;
    int half = lane >> 4, lr = lane & 15;

    // A fragments: aq[ks][vg] = Qgl[m][(2*half+vg)*8 + ks]
    v2f aq[8];
    int m = qt * 16 + lr;
#pragma unroll
    for (int ks = 0; ks < 8; ++ks) {
        int dA = (2 * half) * 8 + ks;
        int dB = (2 * half + 1) * 8 + ks;
        float a0 = 0.f, a1 = 0.f;
        if (m == 0) {
            a0 = glob_l[dA * 8 + h];
            a1 = glob_l[dB * 8 + h];
        } else if (m <= 1024) {
            a0 = ql[((size_t)b * CC + (dA * 8 + h)) * HWW + (m - 1)];
            a1 = ql[((size_t)b * CC + (dB * 8 + h)) * HWW + (m - 1)];
        }
        aq[ks][0] = a0; aq[ks][1] = a1;
    }

    // Pass 1: S = Q K^T * SCALE  -> LDS  (K rows padded to 1440, unconditional b128 loads)
    const float* Kbase = msq + (size_t)b * NKV_PAD * CC + h * DHH + 16 * half;
#pragma unroll 2
    for (int jt = 0; jt < JTILES; ++jt) {
        int j = jt * 16 + lr;
        const float* kp = Kbase + (size_t)j * CC;
        v4f q0 = *(const v4f*)(kp);
        v4f q1 = *(const v4f*)(kp + 4);
        v4f q2 = *(const v4f*)(kp + 8);
        v4f q3 = *(const v4f*)(kp + 12);
        v8f acc = {0.f,0.f,0.f,0.f,0.f,0.f,0.f,0.f};
#pragma unroll
        for (int ks = 0; ks < 8; ++ks) {
            v2f bf;
            bf[0] = (ks < 4) ? q0[ks] : q1[ks - 4];
            bf[1] = (ks < 4) ? q2[ks] : q3[ks - 4];
            acc = wmma4(aq[ks], bf, acc);
        }
        bool inr = (j < NKV);
#pragma unroll
        for (int r = 0; r < 8; ++r)
            sS[(r + 8 * half) * SROW + jt * 16 + lr] = inr ? acc[r] * SCALE_F : -3.0e38f;
    }
    __syncthreads();

    // softmax: lane lr owns row lr; halves split the 1440 columns (float4 scans)
    {
        float* rp = sS + lr * SROW + half * 720;
        float mx = -3.0e38f;
        for (int i = 0; i < 180; ++i) {
            v4f v = *(const v4f*)(rp + i * 4);
            mx = fmaxf(mx, fmaxf(fmaxf(v[0], v[1]), fmaxf(v[2], v[3])));
        }
        mx = fmaxf(mx, __shfl_xor(mx, 16, 32));
        float sum = 0.f;
        for (int i = 0; i < 180; ++i) {
            v4f v = *(const v4f*)(rp + i * 4);
            v4f e;
            e[0] = expf(v[0] - mx); e[1] = expf(v[1] - mx);
            e[2] = expf(v[2] - mx); e[3] = expf(v[3] - mx);
            *(v4f*)(rp + i * 4) = e;
            sum += (e[0] + e[1]) + (e[2] + e[3]);
        }
        sum += __shfl_xor(sum, 16, 32);
        if (half == 0) sDen[lr] = sum;
    }
    __syncthreads();

    // Pass 2: O = P V, slot perm j = jt*16 + k*4 + ks; P from LDS b128, V^T b128
    const float* vtl0 = msvt + ((size_t)b * CC + h * DHH + lr) * NKV_PAD;
    const float* vtl1 = msvt + ((size_t)b * CC + h * DHH + 16 + lr) * NKV_PAD;
    v8f o0 = {0.f,0.f,0.f,0.f,0.f,0.f,0.f,0.f};
    v8f o1 = {0.f,0.f,0.f,0.f,0.f,0.f,0.f,0.f};
#pragma unroll 2
    for (int jt = 0; jt < JTILES; ++jt) {
        int cb = jt * 16 + 8 * half;
        v4f pa = *(const v4f*)(sS + lr * SROW + cb);
        v4f pb = *(const v4f*)(sS + lr * SROW + cb + 4);
        v4f w0 = *(const v4f*)(vtl0 + cb);
        v4f w1 = *(const v4f*)(vtl0 + cb + 4);
        v4f x0 = *(const v4f*)(vtl1 + cb);
        v4f x1 = *(const v4f*)(vtl1 + cb + 4);
#pragma unroll
        for (int ks = 0; ks < 4; ++ks) {
            v2f ap; ap[0] = pa[ks]; ap[1] = pb[ks];
            v2f b0; b0[0] = w0[ks]; b0[1] = w1[ks];
            v2f b1; b1[0] = x0[ks]; b1[1] = x1[ks];
            o0 = wmma4(ap, b0, o0);
            o1 = wmma4(ap, b1, o1);
        }
    }
#pragma unroll
    for (int r = 0; r < 8; ++r) {
        int row = r + 8 * half;
        int gq = qt * 16 + row;
        if (gq >= 1 && gq <= 1024) {
            float inv = 1.0f / sDen[row];
            size_t base = (((size_t)(b * NHH + h)) * 1024 + (gq - 1)) * DHH;
            att_fea[base + lr]      = o0[r] * inv;
            att_fea[base + 16 + lr] = o1[r] * inv;
        }
    }
}

// ---------------- fmap = att_fea @ w_attn, scattered to image layout ----------------
__global__ __launch_bounds__(32) void fmap_kernel(const float* __restrict__ att_fea,
                                                  const float* __restrict__ wattn,
                                                  float* __restrict__ out_img) {
    int blk = blockIdx.x;
    int qt = blk & 63; int bh = blk >> 6;
    int h = bh & 7; int b = bh >> 3;
    int lane = threadIdx.x; int half = lane >> 4, lr = lane & 15;
    const float* A = att_fea + (((size_t)(b * NHH + h)) * 1024) * DHH;
    const float* W = wattn + h * 32 * 32;

    // preload this lane's W columns: w0a[q] = W[16*half+q][lr], w1a[q] = W[16*half+q][16+lr]
    float w0a[16], w1a[16];
#pragma unroll
    for (int q = 0; q < 16; ++q) {
        int d = 16 * half + q;
        w0a[q] = W[d * 32 + lr];
        w1a[q] = W[d * 32 + 16 + lr];
    }
    // A row (contiguous): fea[tok][16*half .. +16) as 4x b128
    const float* ar = A + (size_t)(qt * 16 + lr) * DHH + 16 * half;
    v4f f0 = *(const v4f*)(ar);
    v4f f1 = *(const v4f*)(ar + 4);
    v4f f2 = *(const v4f*)(ar + 8);
    v4f f3 = *(const v4f*)(ar + 12);

    v8f o0 = {0.f,0.f,0.f,0.f,0.f,0.f,0.f,0.f};
    v8f o1 = {0.f,0.f,0.f,0.f,0.f,0.f,0.f,0.f};
#pragma unroll
    for (int ks = 0; ks < 8; ++ks) {
        v2f a;
        a[0] = (ks < 4) ? f0[ks] : f1[ks - 4];
        a[1] = (ks < 4) ? f2[ks] : f3[ks - 4];
        v2f b0; b0[0] = w0a[ks]; b0[1] = w0a[8 + ks];
        v2f b1; b1[0] = w1a[ks]; b1[1] = w1a[8 + ks];
        o0 = wmma4(a, b0, o0);
        o1 = wmma4(a, b1, o1);
    }
#pragma unroll
    for (int r = 0; r < 8; ++r) {
        int tok = qt * 16 + r + 8 * half;
        out_img[(((size_t)b * CC) + (lr * NHH + h)) * HWW + tok]        = o0[r];
        out_img[(((size_t)b * CC) + ((16 + lr) * NHH + h)) * HWW + tok] = o1[r];
    }
}

// ---------------- depthwise 3x3 (pad 1, no bias) ----------------
__global__ __launch_bounds__(256) void dw3_kernel(const float* __restrict__ in,
                                                  const float* __restrict__ w,
                                                  float* __restrict__ out) {
    int idx = blockIdx.x * 256 + threadIdx.x;
    int p = idx & 1023; int c = (idx >> 10) & 255; int b = idx >> 18;
    int y = p >> 5, x = p & 31;
    const float* src = in + ((size_t)b * CC + c) * HWW;
    const float* wk = w + c * 9;
    float acc = 0.f;
#pragma unroll
    for (int dy = 0; dy < 3; ++dy)
#pragma unroll
        for (int dx = 0; dx < 3; ++dx) {
            int yy = y + dy - 1, xx = x + dx - 1;
            if (yy >= 0 && yy < 32 && xx >= 0 && xx < 32)
                acc += wk[dy * 3 + dx] * src[yy * 32 + xx];
        }
    out[idx] = acc;
}

// ---------------- launch ----------------
extern "C" void kernel_launch(void* const* d_in, const int* in_sizes, int n_in,
                              void* d_out, int out_size, void* d_ws, size_t ws_size,
                              hipStream_t stream) {
    const float* h_fea   = (const float*)d_in[0];
    const float* l_fea   = (const float*)d_in[1];
    const float* qv_h_w  = (const float*)d_in[2];
    const float* qv_l_w  = (const float*)d_in[3];
    const float* dconv_w = (const float*)d_in[4];
    const float* dconv_b = (const float*)d_in[5];
    const float* norm_g  = (const float*)d_in[8];
    const float* norm_b  = (const float*)d_in[9];
    const float* glob_l  = (const float*)d_in[13];
    const float* hl_ln_b = (const float*)d_in[19];
    const float* hl_w    = (const float*)d_in[20];
    const float* hl_b    = (const float*)d_in[21];
    const float* out_h_dw = (const float*)d_in[22];
    const float* out_h_pw = (const float*)d_in[23];
    const float* out_l_dw = (const float*)d_in[24];
    const float* out_l_pw = (const float*)d_in[25];

    float* ws = (float*)d_ws;
    float* qvh     = ws + OFF_QVH;
    float* ql      = ws + OFF_QL;
    float* pq      = ws + OFF_PQ;
    float* pv      = ws + OFF_PV;
    float* msq     = ws + OFF_MSQ;
    float* msv     = ws + OFF_MSV;
    float* msvt    = ws + OFF_MSVT;
    float* wattn   = ws + OFF_WATTN;
    float* att_fea = pq;   // reuse
    float* out_img = pv;   // reuse
    float* tmp     = ql;   // reuse

    float* outh = (float*)d_out;
    float* outl = (float*)d_out + (size_t)BB * CC * HWW;

    // projections
    proj_kernel<<<(BB * 512 * HWW) / 256, 256, 0, stream>>>(qv_h_w, h_fea, qvh, 512);
    proj_kernel<<<(BB * CC * HWW) / 256, 256, 0, stream>>>(qv_l_w, l_fea, ql, CC);

    // multiscale pool + residual dwconv + LN (h branch only; l branch is dead code)
    int npool = 2 * BB * CC * NKV;
    pool_kernel<<<(npool + 255) / 256, 256, 0, stream>>>(qvh, pq, pv);
    convms_kernel<<<(npool + 255) / 256, 256, 0, stream>>>(pq, pv, dconv_w, dconv_b, msq, msv);
    ln_kernel<<<2 * BB * NKV, 256, 0, stream>>>(msq, msv, msvt, norm_g, norm_b);
    pad_kernel<<<(BB * 14 * CC + 255) / 256, 256, 0, stream>>>(msq, msvt);

    // constant reweighting matrix (LN over size-1 axis collapses glob_token away)
    wattn_kernel<<<1, 512, 0, stream>>>(hl_ln_b, hl_w, hl_b, wattn);

    // fused WMMA attention
    size_t lds = (16 * SROW + 16) * sizeof(float);
    attn_kernel<<<BB * NHH * QTILES, 32, lds, stream>>>(ql, glob_l, msq, msvt, att_fea);

    // fmap @ w_attn + heads->image
    fmap_kernel<<<BB * NHH * 64, 32, 0, stream>>>(att_fea, wattn, out_img);

    // outl = out_l_pw @ dwconv(out); outh = out_h_pw @ dwconv(outl)
    int nimg = BB * CC * HWW;
    dw3_kernel<<<nimg / 256, 256, 0, stream>>>(out_img, out_l_dw, tmp);
    proj_kernel<<<nimg / 256, 256, 0, stream>>>(out_l_pw, tmp, outl, CC);
    dw3_kernel<<<nimg / 256, 256, 0, stream>>>(outl, out_h_dw, tmp);
    proj_kernel<<<nimg / 256, 256, 0, stream>>>(out_h_pw, tmp, outh, CC);
}